// GraphSageWithSampling_35648228556867
// MI455X (gfx1250) — compile-verified
//
#include <hip/hip_runtime.h>
#include <hip/hip_bf16.h>

#define NODES 50000
#define FEAT  128
#define CDIM  300

typedef __attribute__((ext_vector_type(2))) float v2f;
typedef __attribute__((ext_vector_type(8))) float v8f;

// ---------------------------------------------------------------------------
__global__ void zero_f32(float* __restrict__ p, int n) {
    int i = blockIdx.x * blockDim.x + threadIdx.x;
    if (i < n) p[i] = 0.0f;
}

// ---------------------------------------------------------------------------
__global__ void deg_count(const int* __restrict__ dst, float* __restrict__ deg, int ne) {
    int e = blockIdx.x * blockDim.x + threadIdx.x;
    if (e < ne) atomicAdd(&deg[dst[e]], 1.0f);
}

// ---------------------------------------------------------------------------
// Edge aggregation: hagg[dst] += h[src]. One wave per edge, grid-stride.
// Each lane carries 4 floats (float4 gather -> global_load_b128), then 4 f32
// atomics that resolve in L2 (h and hagg both fit in the 192MB L2).
// ---------------------------------------------------------------------------
__global__ void aggregate(const float* __restrict__ h,
                          const int* __restrict__ src,
                          const int* __restrict__ dst,
                          float* __restrict__ hagg, int ne) {
    int wave   = (blockIdx.x * blockDim.x + threadIdx.x) >> 5;
    int lane   = threadIdx.x & 31;
    int nwaves = (gridDim.x * blockDim.x) >> 5;
    for (int e = wave; e < ne; e += nwaves) {
        int s = src[e];
        int d = dst[e];
        float4 v = ((const float4*)(h + (long)s * FEAT))[lane];
        float* o = hagg + (long)d * FEAT + lane * 4;
        atomicAdd(o + 0, v.x);
        atomicAdd(o + 1, v.y);
        atomicAdd(o + 2, v.z);
        atomicAdd(o + 3, v.w);
    }
}

// ---------------------------------------------------------------------------
// h0 = node_emb[ids+1] + leaky_relu(content @ proj_W.T + proj_b, 0.1)
// One wave computes a full 16x128 output stripe (8 N-tiles): the A fragment
// is loaded once per k-step and reused against 8 B fragments -> 8 WMMAs.
// A-frag (16x4 f32): lanes 0-15: M=lane, K=k+0/k+1; lanes 16-31: M=lane-16,
// K=k+2/k+3. B mirrored. C/D: VGPR r -> M=m0+r+8*half, N=n0+(lane&15).
// ---------------------------------------------------------------------------
__global__ void proj_gemm(const float* __restrict__ content,
                          const float* __restrict__ projW,
                          const float* __restrict__ projB,
                          const float* __restrict__ node_emb,
                          float* __restrict__ h) {
    const int m0   = blockIdx.x * 16;
    const int lane = threadIdx.x;
    const int half = lane >> 4;
    const int l    = lane & 15;

    const float* arow = content + (long)(m0 + l) * CDIM + 2 * half;
    const float* brow[8];
#pragma unroll
    for (int j = 0; j < 8; ++j)
        brow[j] = projW + (long)(j * 16 + l) * CDIM + 2 * half;

    v8f acc[8] = {};
    for (int k = 0; k < CDIM; k += 4) {
        v2f a;
        a.x = arow[k];
        a.y = arow[k + 1];
#pragma unroll
        for (int j = 0; j < 8; ++j) {
            v2f b;
            b.x = brow[j][k];
            b.y = brow[j][k + 1];
            acc[j] = __builtin_amdgcn_wmma_f32_16x16x4_f32(
                false, a, false, b, (short)0, acc[j], false, false);
        }
    }

#pragma unroll
    for (int j = 0; j < 8; ++j) {
        const int n = j * 16 + l;
        const float bias = projB[n];
#pragma unroll
        for (int r = 0; r < 8; ++r) {
            int m = m0 + r + 8 * half;
            float v = acc[j][r] + bias;
            v = (v > 0.0f) ? v : 0.1f * v;               // leaky_relu(0.1)
            v += node_emb[(long)(m + 1) * FEAT + n];     // ids offset by +1
            h[(long)m * FEAT + n] = v;
        }
    }
}

// ---------------------------------------------------------------------------
// Fully fused SAGE layer tail:
//   a1 = (hagg - h) / max(deg-1, 1)            (mean-adjust, on the fly)
//   t  = concat([h, a1]) @ W.T + b             (WMMA, K=256)
//   t  = act ? leaky_relu(t, 0.1) : t
//   out = t / max(||t||_2, 1e-6)               (row norm fused in epilogue)
// One wave owns rows m0..m0+15 across all 128 output cols, so the row-norm
// is 8 local sums + a shfl_xor butterfly over each 16-lane half (rows r and
// r+8 live in separate halves per the C/D layout).
// ---------------------------------------------------------------------------
__global__ void sage_fused(const float* __restrict__ h,
                           const float* __restrict__ hagg,
                           const float* __restrict__ deg,
                           const float* __restrict__ W,    // [128, 256] row-major
                           const float* __restrict__ bias,
                           float* __restrict__ out, int act) {
    const int m0   = blockIdx.x * 16;
    const int lane = threadIdx.x;
    const int half = lane >> 4;
    const int l    = lane & 15;

    // Mean divisor for this lane's A-row (row m0+l in the A fragment).
    const float invA = 1.0f / fmaxf(deg[m0 + l] - 1.0f, 1.0f);

    const float* arow0 = h    + (long)(m0 + l) * FEAT + 2 * half;
    const float* arow1 = hagg + (long)(m0 + l) * FEAT + 2 * half;
    const float* brow[8];
#pragma unroll
    for (int j = 0; j < 8; ++j)
        brow[j] = W + (long)(j * 16 + l) * (2 * FEAT) + 2 * half;

    v8f acc[8] = {};
    for (int k = 0; k < FEAT; k += 4) {
        v2f a0, a1;
        a0.x = arow0[k];
        a0.y = arow0[k + 1];
        a1.x = (arow1[k]     - a0.x) * invA;      // (hagg - h) / max(deg-1,1)
        a1.y = (arow1[k + 1] - a0.y) * invA;
#pragma unroll
        for (int j = 0; j < 8; ++j) {
            v2f b0, b1;
            b0.x = brow[j][k];
            b0.y = brow[j][k + 1];
            b1.x = brow[j][FEAT + k];
            b1.y = brow[j][FEAT + k + 1];
            acc[j] = __builtin_amdgcn_wmma_f32_16x16x4_f32(
                false, a0, false, b0, (short)0, acc[j], false, false);
            acc[j] = __builtin_amdgcn_wmma_f32_16x16x4_f32(
                false, a1, false, b1, (short)0, acc[j], false, false);
        }
    }

    float bn[8];
#pragma unroll
    for (int j = 0; j < 8; ++j) bn[j] = bias[j * 16 + l];

#pragma unroll
    for (int r = 0; r < 8; ++r) {
        const int m = m0 + r + 8 * half;
        float val[8];
        float ss = 0.0f;
#pragma unroll
        for (int j = 0; j < 8; ++j) {
            float v = acc[j][r] + bn[j];
            if (act) v = (v > 0.0f) ? v : 0.1f * v;
            val[j] = v;
            ss += v * v;
        }
        // Row sum of squares: reduce across the 16 lanes of this half.
#pragma unroll
        for (int msk = 8; msk >= 1; msk >>= 1) ss += __shfl_xor(ss, msk, 32);
        const float inv = 1.0f / fmaxf(sqrtf(ss), 1e-6f);
#pragma unroll
        for (int j = 0; j < 8; ++j)
            out[(long)m * FEAT + j * 16 + l] = val[j] * inv;
    }
}

// ---------------------------------------------------------------------------
extern "C" void kernel_launch(void* const* d_in, const int* in_sizes, int n_in,
                              void* d_out, int out_size, void* d_ws, size_t ws_size,
                              hipStream_t stream) {
    const float* node_emb = (const float*)d_in[0];   // (N+1)*128
    const float* content  = (const float*)d_in[1];   // N*300
    const float* projW    = (const float*)d_in[2];   // 128*300
    const float* projB    = (const float*)d_in[3];   // 128
    const float* W1       = (const float*)d_in[4];   // 128*256
    const float* b1       = (const float*)d_in[5];   // 128
    const float* W2       = (const float*)d_in[6];   // 128*256
    const float* b2       = (const float*)d_in[7];   // 128
    const int*   esrc     = (const int*)d_in[8];     // E+N
    const int*   edst     = (const int*)d_in[9];     // E+N
    const int    ne       = in_sizes[8];

    float* hA   = (float*)d_ws;                      // N*128
    float* hB   = hA   + (long)NODES * FEAT;         // N*128
    float* hagg = hB   + (long)NODES * FEAT;         // N*128
    float* deg  = hagg + (long)NODES * FEAT;         // N
    float* out  = (float*)d_out;

    const int NF = NODES * FEAT;                     // 6.4M
    const int gemmBlocks = NODES / 16;               // 3125 waves

    // h0 = node_emb[ids+1] + leaky_relu(content @ proj_W.T + b)
    proj_gemm<<<gemmBlocks, 32, 0, stream>>>(content, projW, projB, node_emb, hA);

    // in-degree (shared by both layers)
    zero_f32<<<(NODES + 255) / 256, 256, 0, stream>>>(deg, NODES);
    deg_count<<<(ne + 255) / 256, 256, 0, stream>>>(edst, deg, ne);

    const int aggBlocks = 8192;                      // 65536 waves, grid-stride

    // ----- layer 1 (leaky_relu) -----
    zero_f32<<<(NF + 255) / 256, 256, 0, stream>>>(hagg, NF);
    aggregate<<<aggBlocks, 256, 0, stream>>>(hA, esrc, edst, hagg, ne);
    sage_fused<<<gemmBlocks, 32, 0, stream>>>(hA, hagg, deg, W1, b1, hB, 1);

    // ----- layer 2 (linear) -----
    zero_f32<<<(NF + 255) / 256, 256, 0, stream>>>(hagg, NF);
    aggregate<<<aggBlocks, 256, 0, stream>>>(hB, esrc, edst, hagg, ne);
    sage_fused<<<gemmBlocks, 32, 0, stream>>>(hB, hagg, deg, W2, b2, out, 0);
}